// GCN_74208444940737
// MI455X (gfx1250) — compile-verified
//
#include <hip/hip_runtime.h>
#include <math.h>

typedef __attribute__((ext_vector_type(2))) float v2f;
typedef __attribute__((ext_vector_type(8))) float v8f;

// ------------------------------------------------------------------
// init: zero the accumulators used this call (harness does not re-poison)
// ------------------------------------------------------------------
__global__ void k_init(int* in_deg, int* out_deg, float* gsum, int* gcnt,
                       int N, int GH, int G) {
  int i = blockIdx.x * blockDim.x + threadIdx.x;
  if (i < N) { in_deg[i] = 0; out_deg[i] = 0; }
  if (i < GH) gsum[i] = 0.0f;
  if (i < G)  gcnt[i] = 0;
}

// ------------------------------------------------------------------
// degree histogram (int atomics, 2 per edge)
// ------------------------------------------------------------------
__global__ void k_degrees(const int* __restrict__ esrc, const int* __restrict__ edst,
                          int* out_deg, int* in_deg, int E) {
  int e = blockIdx.x * blockDim.x + threadIdx.x;
  if (e >= E) return;
  atomicAdd(&out_deg[esrc[e]], 1);
  atomicAdd(&in_deg[edst[e]], 1);
}

// ------------------------------------------------------------------
// single-block exclusive scan of in-degree -> CSR row offsets (+cursor copy)
// ------------------------------------------------------------------
__global__ void k_scan(const int* __restrict__ in, int* row_off, int* cursor, int n) {
  __shared__ int sm[1024];
  __shared__ int carry_sm;
  if (threadIdx.x == 0) carry_sm = 0;
  __syncthreads();
  for (int base = 0; base < n; base += 1024) {
    int i = base + (int)threadIdx.x;
    int v = (i < n) ? in[i] : 0;
    sm[threadIdx.x] = v;
    __syncthreads();
    #pragma unroll
    for (int off = 1; off < 1024; off <<= 1) {
      int t = (threadIdx.x >= (unsigned)off) ? sm[threadIdx.x - off] : 0;
      __syncthreads();
      sm[threadIdx.x] += t;
      __syncthreads();
    }
    int incl = sm[threadIdx.x];
    int carry = carry_sm;
    if (i < n) { int ex = carry + incl - v; row_off[i] = ex; cursor[i] = ex; }
    __syncthreads();
    if (threadIdx.x == 1023) carry_sm = carry + incl;
    __syncthreads();
  }
  if (threadIdx.x == 0) row_off[n] = carry_sm;
}

// ------------------------------------------------------------------
// CSR bucket fill (dst-keyed, lists src)
// ------------------------------------------------------------------
__global__ void k_fill(const int* __restrict__ esrc, const int* __restrict__ edst,
                       int* cursor, int* csr_src, int E) {
  int e = blockIdx.x * blockDim.x + threadIdx.x;
  if (e >= E) return;
  int d = edst[e];
  int pos = atomicAdd(&cursor[d], 1);
  csr_src[pos] = esrc[e];
}

// ------------------------------------------------------------------
// per-node degree scales + per-graph node counts
// ------------------------------------------------------------------
__global__ void k_nodeprep(const int* __restrict__ out_deg, const int* __restrict__ in_deg,
                           const int* __restrict__ gid,
                           float* sout, float* sin_, int* gcnt, int N) {
  int i = blockIdx.x * blockDim.x + threadIdx.x;
  if (i >= N) return;
  float od = (float)out_deg[i]; if (od < 1.0f) od = 1.0f;
  float idg = (float)in_deg[i]; if (idg < 1.0f) idg = 1.0f;
  sout[i] = 1.0f / sqrtf(od);
  sin_[i] = 1.0f / sqrtf(idg);
  atomicAdd(&gcnt[gid[i]], 1);
}

// ------------------------------------------------------------------
// layer-1 aggregation via CSR (2 features; thread-per-node, no atomics)
// ------------------------------------------------------------------
__global__ void k_l1_gather(const int* __restrict__ row_off, const int* __restrict__ csr_src,
                            const float* __restrict__ aTime, const float* __restrict__ aType,
                            const float* __restrict__ sout, const float* __restrict__ sin_,
                            float* agg1, int N) {
  int i = blockIdx.x * blockDim.x + threadIdx.x;
  if (i >= N) return;
  float a0 = 0.0f, a1 = 0.0f;
  int s0 = row_off[i], s1 = row_off[i + 1];
  for (int e = s0; e < s1; ++e) {
    int s = csr_src[e];
    float w = sout[s];
    a0 += aTime[s] * w;
    a1 += aType[s] * w;
  }
  float si = sin_[i];
  agg1[2 * i]     = a0 * si;
  agg1[2 * i + 1] = a1 * si;
}

// ------------------------------------------------------------------
// layer-1 dense [N,2]@[2,H] + bias + relu; pre-multiply out_deg^-1/2
// ------------------------------------------------------------------
__global__ void k_l1_dense(const float* __restrict__ agg1, const float* __restrict__ W1,
                           const float* __restrict__ b1, const float* __restrict__ sout,
                           float* h1, int N, int H) {
  int n = blockIdx.x;
  int j = threadIdx.x;
  float a0 = agg1[2 * n], a1 = agg1[2 * n + 1];
  float h = fmaf(a0, W1[j], fmaf(a1, W1[H + j], b1[j]));
  h = h > 0.0f ? h : 0.0f;
  h1[n * H + j] = h * sout[n];
}

// ------------------------------------------------------------------
// layer-2 aggregation: wave-per-node CSR gather (wave32, 8 floats/lane)
// ------------------------------------------------------------------
__global__ void k_l2_gather(const int* __restrict__ row_off, const int* __restrict__ csr_src,
                            const float* __restrict__ h1, const float* __restrict__ sin_,
                            float* __restrict__ agg2, int N) {
  const int H = 256;
  int node = blockIdx.x * 8 + (int)(threadIdx.x >> 5);
  int lane = threadIdx.x & 31;
  if (node >= N) return;
  int s0 = row_off[node], s1 = row_off[node + 1];
  float4 a0 = make_float4(0.f, 0.f, 0.f, 0.f);
  float4 a1 = make_float4(0.f, 0.f, 0.f, 0.f);
  for (int e = s0; e < s1; ++e) {
    int s = csr_src[e];
    const float4* p = (const float4*)(h1 + s * H) + lane * 2;
    float4 x0 = p[0], x1 = p[1];
    a0.x += x0.x; a0.y += x0.y; a0.z += x0.z; a0.w += x0.w;
    a1.x += x1.x; a1.y += x1.y; a1.z += x1.z; a1.w += x1.w;
  }
  float si = sin_[node];
  float4* q = (float4*)(agg2 + node * H) + lane * 2;
  q[0] = make_float4(a0.x * si, a0.y * si, a0.z * si, a0.w * si);
  q[1] = make_float4(a1.x * si, a1.y * si, a1.z * si, a1.w * si);
}

// ------------------------------------------------------------------
// layer-2 dense via V_WMMA_F32_16X16X4_F32.
// 256 threads = 8 waves; block computes a 64-row x 256-col panel:
//   wave w -> columns [w*32, w*32+32), all 64 rows (4 M-tiles x 2 N-tiles,
//   8 v8f accumulators). W2 read once per block -> W2 L2 traffic /4 vs
//   16-row blocks. A panel + graph_id staged in LDS.
// Epilogue: bias + relu + per-graph mean-pool scatter, with a branch-free
// fast path for full blocks (guards only in the single tail block).
// ------------------------------------------------------------------
#define ROWS 64

template <bool GUARD>
__device__ __forceinline__ void l2_epilogue(const v8f (&c)[4][2], const float* __restrict__ b2,
                                            const int* gidS, float* gsum,
                                            int m0, int wave, int half, int l, int N) {
  const int H = 256;
  float bias[2];
  #pragma unroll
  for (int nt = 0; nt < 2; ++nt) bias[nt] = b2[wave * 32 + nt * 16 + l];
  #pragma unroll
  for (int mt = 0; mt < 4; ++mt) {
    #pragma unroll
    for (int nt = 0; nt < 2; ++nt) {
      int ncol = wave * 32 + nt * 16 + l;
      #pragma unroll
      for (int r = 0; r < 8; ++r) {
        int rr = mt * 16 + r + half * 8;          // D layout: vgpr r = row r (+8 hi half)
        if (GUARD && (m0 + rr) >= N) continue;
        float v = c[mt][nt][r] + bias[nt];
        v = v > 0.0f ? v : 0.0f;
        atomicAdd(&gsum[gidS[rr] * H + ncol], v);
      }
    }
  }
}

__global__ __launch_bounds__(256)
void k_layer2_wmma_pool(const float* __restrict__ agg2, const float* __restrict__ W2,
                        const float* __restrict__ b2, const int* __restrict__ gid,
                        float* gsum, int N) {
  const int H = 256;
  __shared__ float As[ROWS][260];   // pad 260: bank = (4*row + k) % 64, conflict-free
  __shared__ int gidS[ROWS];
  int m0 = blockIdx.x * ROWS;

  // cooperative stage: A panel (float4 path) + graph ids
  for (int idx = threadIdx.x; idx < ROWS * 64; idx += 256) {
    int r = idx >> 6;
    int c4 = (idx & 63) * 4;
    int m = m0 + r; if (m >= N) m = N - 1;       // clamp; tail stores are guarded
    float4 v = *(const float4*)(agg2 + m * H + c4);
    *(float4*)&As[r][c4] = v;
  }
  if (threadIdx.x < ROWS) {
    int m = m0 + (int)threadIdx.x; if (m >= N) m = N - 1;
    gidS[threadIdx.x] = gid[m];
  }
  __syncthreads();

  int lane = threadIdx.x & 31;
  int wave = (int)(threadIdx.x >> 5);   // 0..7 -> 32 columns each
  int half = lane >> 4;                 // K pair selector (K+0,1 vs K+2,3)
  int l    = lane & 15;
  int colbase = wave * 32 + l;

  v8f c[4][2] = {};
  for (int k0 = 0; k0 < H; k0 += 4) {
    int kb = k0 + half * 2;
    // A 16x4 f32 layout: lanes 0-15 rows M with K=k0..k0+1; lanes 16-31 K=k0+2..k0+3
    v2f a[4];
    #pragma unroll
    for (int mt = 0; mt < 4; ++mt)
      a[mt] = *(const v2f*)&As[mt * 16 + l][kb];
    const float* w0 = W2 + kb * H;
    const float* w1 = w0 + H;
    v2f b[2];
    #pragma unroll
    for (int nt = 0; nt < 2; ++nt) {
      b[nt].x = w0[colbase + 16 * nt];
      b[nt].y = w1[colbase + 16 * nt];
    }
    #pragma unroll
    for (int mt = 0; mt < 4; ++mt) {
      #pragma unroll
      for (int nt = 0; nt < 2; ++nt) {
        c[mt][nt] = __builtin_amdgcn_wmma_f32_16x16x4_f32(false, a[mt], false, b[nt],
                                                          (short)0, c[mt][nt], false, false);
      }
    }
  }

  if (m0 + ROWS <= N) {
    l2_epilogue<false>(c, b2, gidS, gsum, m0, wave, half, l, N);
  } else {
    l2_epilogue<true>(c, b2, gidS, gsum, m0, wave, half, l, N);
  }
}

// ------------------------------------------------------------------
// head: out[g,o] = (sum_k gsum[g,k] * Wc[k,o]) / max(cnt,1) + bc[o]
// ------------------------------------------------------------------
__global__ void k_finalize(const float* __restrict__ gsum, const int* __restrict__ gcnt,
                           const float* __restrict__ Wc, const float* __restrict__ bc,
                           float* out, int G, int H) {
  int i = blockIdx.x * blockDim.x + threadIdx.x;
  if (i >= G * 2) return;
  int g = i >> 1, o = i & 1;
  float cnt = (float)gcnt[g];
  if (cnt < 1.0f) cnt = 1.0f;
  float s = 0.0f;
  for (int k = 0; k < H; ++k) s += gsum[g * H + k] * Wc[k * 2 + o];
  out[i] = s / cnt + bc[o];
}

extern "C" void kernel_launch(void* const* d_in, const int* in_sizes, int n_in,
                              void* d_out, int out_size, void* d_ws, size_t ws_size,
                              hipStream_t stream) {
  const float* aTime = (const float*)d_in[0];
  const float* aType = (const float*)d_in[1];
  const int*   esrc  = (const int*)d_in[2];
  const int*   edst  = (const int*)d_in[3];
  const int*   gid   = (const int*)d_in[4];
  // d_in[5] = n_graphs device scalar; G derived from out_size instead
  const float* W1    = (const float*)d_in[6];
  const float* b1    = (const float*)d_in[7];
  const float* W2    = (const float*)d_in[8];
  const float* b2    = (const float*)d_in[9];
  const float* Wc    = (const float*)d_in[10];
  const float* bc    = (const float*)d_in[11];
  float* out = (float*)d_out;

  const int N = in_sizes[0];
  const int E = in_sizes[2];
  const int H = in_sizes[7];     // 256
  const int G = out_size / 2;
  (void)n_in; (void)ws_size;

  char* p = (char*)d_ws;
  auto alloc = [&p](size_t bytes) -> void* {
    void* r = (void*)p;
    p += (bytes + 255) & ~(size_t)255;
    return r;
  };
  int*   in_deg  = (int*)alloc((size_t)N * 4);
  int*   out_deg = (int*)alloc((size_t)N * 4);
  int*   row_off = (int*)alloc((size_t)(N + 1) * 4);
  int*   cursor  = (int*)alloc((size_t)N * 4);
  int*   csr_src = (int*)alloc((size_t)E * 4);
  float* sout    = (float*)alloc((size_t)N * 4);
  float* sin_    = (float*)alloc((size_t)N * 4);
  float* agg1    = (float*)alloc((size_t)N * 2 * 4);
  float* h1      = (float*)alloc((size_t)N * H * 4);
  float* agg2    = (float*)alloc((size_t)N * H * 4);
  float* gsum    = (float*)alloc((size_t)G * H * 4);
  int*   gcnt    = (int*)alloc((size_t)G * 4);

  int initN = N > G * H ? N : G * H;
  k_init<<<(initN + 255) / 256, 256, 0, stream>>>(in_deg, out_deg, gsum, gcnt, N, G * H, G);
  k_degrees<<<(E + 255) / 256, 256, 0, stream>>>(esrc, edst, out_deg, in_deg, E);
  k_scan<<<1, 1024, 0, stream>>>(in_deg, row_off, cursor, N);
  k_fill<<<(E + 255) / 256, 256, 0, stream>>>(esrc, edst, cursor, csr_src, E);
  k_nodeprep<<<(N + 255) / 256, 256, 0, stream>>>(out_deg, in_deg, gid, sout, sin_, gcnt, N);
  k_l1_gather<<<(N + 255) / 256, 256, 0, stream>>>(row_off, csr_src, aTime, aType, sout, sin_, agg1, N);
  k_l1_dense<<<N, H, 0, stream>>>(agg1, W1, b1, sout, h1, N, H);
  k_l2_gather<<<(N + 7) / 8, 256, 0, stream>>>(row_off, csr_src, h1, sin_, agg2, N);
  k_layer2_wmma_pool<<<(N + ROWS - 1) / ROWS, 256, 0, stream>>>(agg2, W2, b2, gid, gsum, N);
  k_finalize<<<(G * 2 + 255) / 256, 256, 0, stream>>>(gsum, gcnt, Wc, bc, out, G, H);
}